// GCNConvLayer_SparseAdj_20650202759168
// MI455X (gfx1250) — compile-verified
//
#include <hip/hip_runtime.h>

#define N_NODES 10000
#define N_EDGES 640000
#define D 128

typedef __attribute__((ext_vector_type(2))) float v2f;
typedef __attribute__((ext_vector_type(8))) float v8f;

// Force the native CDNA5 f32 atomic (no CAS-loop fallback).
__device__ __forceinline__ void atomic_add_f32(float* p, float v) {
  asm volatile("global_atomic_add_f32 %0, %1, off" : : "v"(p), "v"(v) : "memory");
}

// ---------------------------------------------------------------------------
// Kernel 1: agg = nfeat (self-loop contribution), deg = 1.0 (self-loop count)
// ---------------------------------------------------------------------------
__global__ void gcn_init(const float* __restrict__ nfeat,
                         float* __restrict__ agg,
                         float* __restrict__ deg) {
  int i = blockIdx.x * blockDim.x + threadIdx.x;
  if (i < N_NODES * D) agg[i] = nfeat[i];
  if (i < N_NODES) deg[i] = 1.0f;
}

// ---------------------------------------------------------------------------
// Kernel 2: one wave32 per edge.
//   agg[src] += nfeat[dst]   (128 floats: 4 per lane, float4 load + 4 atomics)
//   deg[dst] += 1            (lane 0)
// ---------------------------------------------------------------------------
__global__ void gcn_scatter(const float* __restrict__ nfeat,
                            const long long* __restrict__ edges,
                            float* __restrict__ agg,
                            float* __restrict__ deg) {
  int wave = (blockIdx.x * blockDim.x + threadIdx.x) >> 5;
  int lane = threadIdx.x & 31;
  if (wave >= N_EDGES) return;
  long long s = edges[wave];             // src row of edges[2, E]
  long long t = edges[N_EDGES + wave];   // dst row
  const float4 f = *(const float4*)(nfeat + (size_t)t * D + lane * 4);
  float* dp = agg + (size_t)s * D + lane * 4;
  atomic_add_f32(dp + 0, f.x);
  atomic_add_f32(dp + 1, f.y);
  atomic_add_f32(dp + 2, f.z);
  atomic_add_f32(dp + 3, f.w);
  if (lane == 0) atomic_add_f32(deg + t, 1.0f);
}

// ---------------------------------------------------------------------------
// Kernel 3: out = (agg/deg) @ W^T + b via V_WMMA_F32_16X16X4_F32.
// One wave per 16x16 tile: 625 M-tiles x 8 N-tiles = 5000 waves.
// A[r][k]    = agg[m0+r][k] * (1/deg[m0+r])
// B[k][n]    = W[n0+n][k]            (i.e. W^T, contiguous along k -> float2)
// D VGPR v   -> out[m0 + v + 8*half][n0 + r]
// ---------------------------------------------------------------------------
__global__ void gcn_gemm(const float* __restrict__ agg,
                         const float* __restrict__ deg,
                         const float* __restrict__ W,
                         const float* __restrict__ bias,
                         float* __restrict__ out) {
  int waveInBlock = threadIdx.x >> 5;
  int lane = threadIdx.x & 31;
  int tile = blockIdx.x * (blockDim.x >> 5) + waveInBlock;  // 0..4999
  int m0 = (tile >> 3) * 16;   // 0..9984
  int n0 = (tile & 7) * 16;    // 0..112
  int r = lane & 15;
  int half = lane >> 4;

  float invd = 1.0f / deg[m0 + r];
  const float* arow = agg + (size_t)(m0 + r) * D;
  const float* brow = W + (size_t)(n0 + r) * D;

  v8f c = {0.f, 0.f, 0.f, 0.f, 0.f, 0.f, 0.f, 0.f};
#pragma unroll
  for (int k = 0; k < D; k += 4) {
    v2f a = *(const v2f*)(arow + k + 2 * half);
    a *= invd;                                   // fold rst = agg/deg into A
    v2f b = *(const v2f*)(brow + k + 2 * half);
    c = __builtin_amdgcn_wmma_f32_16x16x4_f32(
        /*neg_a=*/false, a, /*neg_b=*/false, b,
        /*c_mod=*/(short)0, c, /*reuse_a=*/false, /*reuse_b=*/false);
  }

  float bb = bias[n0 + r];
#pragma unroll
  for (int v = 0; v < 8; ++v) {
    out[(size_t)(m0 + v + 8 * half) * D + (n0 + r)] = c[v] + bb;
  }
}

// ---------------------------------------------------------------------------
extern "C" void kernel_launch(void* const* d_in, const int* in_sizes, int n_in,
                              void* d_out, int out_size, void* d_ws, size_t ws_size,
                              hipStream_t stream) {
  (void)in_sizes; (void)n_in; (void)out_size; (void)ws_size;
  const float* nfeat = (const float*)d_in[0];
  // d_in[1] = efeat (unused by the reference)
  const long long* edges = (const long long*)d_in[2];
  const float* W = (const float*)d_in[3];
  const float* bias = (const float*)d_in[4];
  float* out = (float*)d_out;

  float* agg = (float*)d_ws;                    // N_NODES*D floats (5.12 MB)
  float* deg = agg + (size_t)N_NODES * D;       // N_NODES floats

  gcn_init<<<(N_NODES * D + 255) / 256, 256, 0, stream>>>(nfeat, agg, deg);
  gcn_scatter<<<N_EDGES / 8, 256, 0, stream>>>(nfeat, edges, agg, deg);  // 8 waves/block
  gcn_gemm<<<625, 256, 0, stream>>>(agg, deg, W, bias, out);             // 5000 waves
}